// LossEvaluator_73701638799800
// MI455X (gfx1250) — compile-verified
//
#include <hip/hip_runtime.h>
#include <cstdint>
#include <cstddef>

#define S_ 128
#define T_ 128
#define B_ 64
#define C_ 8
#define LEV 20.0f

typedef float v2f __attribute__((ext_vector_type(2)));
typedef float v8f __attribute__((ext_vector_type(8)));
typedef int   v4i __attribute__((ext_vector_type(4)));

#if defined(__HIP_DEVICE_COMPILE__) && __has_builtin(__builtin_amdgcn_wmma_f32_16x16x4_f32)
#define HAVE_WMMA 1
#endif
#if defined(__HIP_DEVICE_COMPILE__) && __has_builtin(__builtin_amdgcn_global_load_async_to_lds_b128)
#define HAVE_ASYNC 1
#endif

#if HAVE_ASYNC
typedef __attribute__((address_space(1))) v4i* gv4i_p;   // global
typedef __attribute__((address_space(3))) v4i* lv4i_p;   // LDS
#endif

// ---------------- fast math (hardware transcendentals) ----------------
__device__ __forceinline__ float fast_log(float x) {     // natural log via v_log_f32
#if defined(__HIP_DEVICE_COMPILE__) && __has_builtin(__builtin_amdgcn_logf)
  return __builtin_amdgcn_logf(x) * 0.69314718055994531f;  // log(1)=0 exactly
#else
  return __logf(x);
#endif
}
__device__ __forceinline__ float fast_rcp(float x) {     // v_rcp_f32
#if defined(__HIP_DEVICE_COMPILE__) && __has_builtin(__builtin_amdgcn_rcpf)
  return __builtin_amdgcn_rcpf(x);
#else
  return 1.0f / x;
#endif
}

// ---------------- cross-lane helpers (VALU-only where available) ----------------
#define DPP8SEL(a,b,c,d,e,f,g,h) \
  ((a)|((b)<<3)|((c)<<6)|((d)<<9)|((e)<<12)|((f)<<15)|((g)<<18)|((h)<<21))

template <int SEL>
__device__ __forceinline__ float fdpp8(float x) {
#if defined(__HIP_DEVICE_COMPILE__) && __has_builtin(__builtin_amdgcn_mov_dpp8)
  return __int_as_float(__builtin_amdgcn_mov_dpp8(__float_as_int(x), SEL));
#else
  return 0.0f;  // unused on host pass
#endif
}
#define XOR1_SEL DPP8SEL(1,0,3,2,5,4,7,6)
#define XOR2_SEL DPP8SEL(2,3,0,1,6,7,4,5)
#define XOR4_SEL DPP8SEL(4,5,6,7,0,1,2,3)

__device__ __forceinline__ float fxor8(float x) {   // swap 8-lane halves of each row16
#if defined(__HIP_DEVICE_COMPILE__) && __has_builtin(__builtin_amdgcn_update_dpp)
  return __int_as_float(__builtin_amdgcn_update_dpp(
      0, __float_as_int(x), 0x128 /*row_ror:8*/, 0xf, 0xf, true));
#else
  return __shfl_xor(x, 8, 32);
#endif
}

__device__ __forceinline__ float fxor16(float x) {  // swap 16-lane halves of wave32
#if defined(__HIP_DEVICE_COMPILE__) && __has_builtin(__builtin_amdgcn_permlanex16)
  unsigned i = (unsigned)__float_as_int(x);
  return __int_as_float((int)__builtin_amdgcn_permlanex16(
      i, i, 0x76543210u, 0xfedcba98u, false, false));
#else
  return __shfl_xor(x, 16, 32);
#endif
}

__device__ __forceinline__ float bcast_lane(float x, int l) {
#if defined(__HIP_DEVICE_COMPILE__) && __has_builtin(__builtin_amdgcn_readlane)
  return __int_as_float(__builtin_amdgcn_readlane(__float_as_int(x), l));
#else
  return __shfl(x, l, 32);
#endif
}

// ---------------- JAX-exact Threefry-2x32 ----------------
__device__ __forceinline__ void tf2x32(unsigned k0, unsigned k1,
                                       unsigned x0, unsigned x1,
                                       unsigned& r0, unsigned& r1) {
  const unsigned ks2 = 0x1BD11BDAu ^ k0 ^ k1;
  x0 += k0; x1 += k1;
#define TFR(r) { x0 += x1; x1 = (x1 << (r)) | (x1 >> (32 - (r))); x1 ^= x0; }
  TFR(13) TFR(15) TFR(26) TFR(6)   x0 += k1;  x1 += ks2 + 1u;
  TFR(17) TFR(29) TFR(16) TFR(24)  x0 += ks2; x1 += k0  + 2u;
  TFR(13) TFR(15) TFR(26) TFR(6)   x0 += k0;  x1 += k1  + 3u;
  TFR(17) TFR(29) TFR(16) TFR(24)  x0 += k1;  x1 += ks2 + 4u;
  TFR(13) TFR(15) TFR(26) TFR(6)   x0 += ks2; x1 += k0  + 5u;
#undef TFR
  r0 = x0; r1 = x1;
}

__device__ __forceinline__ float u01(unsigned b) {
  return __uint_as_float((b >> 9) | 0x3f800000u) - 1.0f;  // exact JAX uniform [0,1)
}

// keys = split(key(1234), 128); per step: ke,kt = split(keys[t])
__global__ void tf_init_keys_kernel(unsigned* __restrict__ ws) {
  const unsigned t = threadIdx.x;
  if (t >= T_) return;
  unsigned a0, a1, b0, b1, k0t, k1t;
  if (t < 64) {
    tf2x32(0u, 1234u, 2u*t,        2u*t + 128u, a0, a1); k0t = a0;
    tf2x32(0u, 1234u, 2u*t + 1u,   2u*t + 129u, b0, b1); k1t = b0;
  } else {
    tf2x32(0u, 1234u, 2u*t - 128u, 2u*t,        a0, a1); k0t = a1;
    tf2x32(0u, 1234u, 2u*t - 127u, 2u*t + 1u,   b0, b1); k1t = b1;
  }
  unsigned e0, e1, f0, f1;
  tf2x32(k0t, k1t, 0u, 2u, e0, e1);
  tf2x32(k0t, k1t, 1u, 3u, f0, f1);
  ws[4u*t + 0u] = e0;  // ke0
  ws[4u*t + 1u] = f0;  // ke1
  ws[4u*t + 2u] = e1;  // kt0
  ws[4u*t + 3u] = f1;  // kt1
}

// ---------------- main simulation: 1 block per b, thread = s ----------------
__global__ __launch_bounds__(S_) void
sim_kernel(const float* __restrict__ probs, const float* __restrict__ prices,
           const unsigned* __restrict__ keys, float* __restrict__ out)
{
  __shared__ float    stage[2][48];     // per-step probs(32)+prices(16), double buffered
  __shared__ unsigned keyl[T_][4];      // per-step ke0,ke1,kt0,kt1
  __shared__ unsigned ex[S_][C_];       // threefry half exchange between s and s+64
  __shared__ float    part[4][16];      // per-wave cost partials (B-matrix for WMMA)

  const int tid  = threadIdx.x;
  const int b    = blockIdx.x;
  const int wave = tid >> 5;
  const int lane = tid & 31;

  for (int i = tid; i < T_ * 4; i += S_) keyl[i >> 2][i & 3] = keys[i];
  if (tid < 32) stage[0][tid]      = probs [(size_t)b * C_ * 4 + tid];
  if (tid < 16) stage[0][32 + tid] = prices[(size_t)b * C_ * 2 + tid];
  if (tid < 64) ((float*)part)[tid] = 0.0f;
  __syncthreads();

  unsigned open_bits = 0u, type_bits = 0u;
  float cash = 1.0f, cash_lp = 0.0f, loss = 0.0f;
  bool bankrupt = false;
  float ipv[C_], t00[C_], t10[C_], pcel[C_], ipvl[C_];
#pragma unroll
  for (int c = 0; c < C_; ++c) { ipv[c] = t00[c] = t10[c] = pcel[c] = ipvl[c] = 0.0f; }

  for (int t = 0; t < T_; ++t) {
    const int cur = t & 1, nxt = cur ^ 1;
    const float* stg = stage[cur];

    // ---- prefetch step t+1 operands (async global -> LDS, double buffer) ----
#if HAVE_ASYNC
    if ((t + 1) < T_ && tid < 12) {
      const float* g; float* l;
      if (tid < 8) { g = probs  + ((size_t)(t + 1) * B_ + b) * C_ * 4 + tid * 4;
                     l = &stage[nxt][tid * 4]; }
      else         { g = prices + ((size_t)(t + 1) * B_ + b) * C_ * 2 + (tid - 8) * 4;
                     l = &stage[nxt][32 + (tid - 8) * 4]; }
      __builtin_amdgcn_global_load_async_to_lds_b128((gv4i_p)g, (lv4i_p)l, 0, 0);
    }
#else
    float pf = 0.0f;
    if ((t + 1) < T_ && tid < 48) {
      pf = (tid < 32) ? probs [((size_t)(t + 1) * B_ + b) * C_ * 4 + tid]
                      : prices[((size_t)(t + 1) * B_ + b) * C_ * 2 + (tid - 32)];
    }
#endif

    // ---- PRNG: 8 threefry hashes/thread, halves shared with partner s^64 ----
    const unsigned ke0 = keyl[t][0], ke1 = keyl[t][1];
    const unsigned kt0 = keyl[t][2], kt1 = keyl[t][3];
    const unsigned base_fl = ((unsigned)tid * B_ + (unsigned)b) * C_;  // s*512+b*8
    unsigned mine[C_];
#pragma unroll
    for (int c = 0; c < C_; ++c) {
      unsigned h0, h1;
      const unsigned fl = base_fl + (unsigned)c;
      if (tid < 64) { tf2x32(ke0, ke1, fl, fl + 32768u, h0, h1);  // event hash
                      mine[c] = h0; ex[tid + 64][c] = h1; }
      else          { tf2x32(kt0, kt1, fl - 32768u, fl, h0, h1);  // type hash
                      mine[c] = h1; ex[tid - 64][c] = h0; }
    }
    __syncthreads();  // barrier 1
    unsigned ue[C_], ut[C_];
#pragma unroll
    for (int c = 0; c < C_; ++c) {
      const unsigned o = ex[tid][c];
      ue[c] = (tid < 64) ? mine[c] : o;
      ut[c] = (tid < 64) ? o : mine[c];
    }

    // ---- phase 1: mark-to-market open positions, bankruptcy check ----
    float pos_pl[C_]; float tot = 0.0f;
#pragma unroll
    for (int c = 0; c < C_; ++c) {
      const bool  ob  = (open_bits >> c) & 1u;
      const bool  ty  = (type_bits >> c) & 1u;
      const float pbt = ty ? stg[32 + 2*c + 1] : stg[32 + 2*c + 0];
      const float t11 = (ty ? 1.0f : -1.0f) * fast_rcp(pbt);
      const float pl  = ipv[c] * t00[c] * (t10[c] + t11);
      pos_pl[c] = ob ? pl : 0.0f;
      tot += ob ? (ipv[c] + pl) : 0.0f;
    }
    const float portfolio = cash + tot;
    bankrupt = (open_bits != 0u) ? (portfolio <= 0.0f) : bankrupt;

    // ---- phase 2: events, log-probs, opens/closes (branchless) ----
    unsigned new_open = 0u, new_close = 0u;
    float cost[C_];
#pragma unroll
    for (int c = 0; c < C_; ++c) {
      const bool  ob = (open_bits >> c) & 1u;
      const float pO = stg[4*c + 0];
      const float pC = stg[4*c + 1];
      const float pS = stg[4*c + 2];
      const float pe = ob ? (bankrupt ? 1.0f : pC) : (bankrupt ? 0.0f : pO);
      const bool  ev = u01(ue[c]) < pe;
      const float pcl  = fminf(fmaxf(pe, 1e-7f), 1.0f - 1e-7f);
      const float elog = fast_log(ev ? pcl : 1.0f - pcl);     // one log, selected input
      const bool opening = (!ob) && ev;
      const bool closing = ob && ev;
      open_bits ^= ((unsigned)ev) << c;
      const bool ns = u01(ut[c]) < pS;
      const float psc  = fminf(fmaxf(pS, 1e-7f), 1.0f - 1e-7f);
      const float slog = fast_log(opening ? (ns ? psc : 1.0f - psc) : 1.0f);  // 0 if !opening
      pcel[c] += elog + slog;
      const unsigned m = ((unsigned)opening) << c;
      type_bits = (type_bits & ~m) | ((((unsigned)ns) << c) & m);
      const float po = ns ? stg[32 + 2*c + 1] : stg[32 + 2*c + 0];
      t00[c] = opening ? (LEV * po) : t00[c];
      t10[c] = opening ? (((1.0f / LEV) + (ns ? -1.0f : 1.0f)) * fast_rcp(po)) : t10[c];
      cost[c] = closing ? pos_pl[c] : 0.0f;
      new_open  |= (unsigned)opening << c;
      new_close |= (unsigned)closing << c;
    }

    // ---- cross-S reduction: VALU-only lane exchanges, then LDS partials ----
    float v[C_];
#pragma unroll
    for (int c = 0; c < C_; ++c) v[c] = cost[c] + fxor16(cost[c]);
#pragma unroll
    for (int c = 0; c < C_; ++c) v[c] += fxor8(v[c]);
    // fold xor4: lanes with bit4=0 own cols 0..3, bit4=1 own cols 4..7
    const bool b4 = (lane & 4) != 0, b2 = (lane & 2) != 0, b1 = (lane & 1) != 0;
    float w[4];
#pragma unroll
    for (int j = 0; j < 4; ++j) {
      const float pack = b4 ? v[j] : v[j + 4];
      const float keep = b4 ? v[j + 4] : v[j];
      w[j] = keep + fdpp8<XOR4_SEL>(pack);
    }
    float y[2];
#pragma unroll
    for (int j = 0; j < 2; ++j) {
      const float pack = b2 ? w[j] : w[j + 2];
      const float keep = b2 ? w[j + 2] : w[j];
      y[j] = keep + fdpp8<XOR2_SEL>(pack);
    }
    {
      const float pack = b1 ? y[0] : y[1];
      const float keep = b1 ? y[1] : y[0];
      const float fin  = keep + fdpp8<XOR1_SEL>(pack);  // lane l holds wave-sum of col l&7
      if (lane < 8) part[wave][lane] = fin;
    }

    // make prefetched step-(t+1) data LDS-visible before the barrier
#if HAVE_ASYNC
#if __has_builtin(__builtin_amdgcn_s_wait_asynccnt)
    __builtin_amdgcn_s_wait_asynccnt(0);
#else
    asm volatile("s_wait_asynccnt 0" ::: "memory");
#endif
#else
    if ((t + 1) < T_ && tid < 48) stage[nxt][tid] = pf;
#endif
    __syncthreads();  // barrier 2

    // ---- finish reduction with one f32 WMMA: ones(16x4) x part(4x16) ----
    float baseline[C_];
#if HAVE_WMMA
    {
      v2f am; am[0] = 1.0f; am[1] = 1.0f;          // A = ones
      const int hi = lane >> 4, col = lane & 15;
      v2f bm;                                      // B[k][n] = part[k][n]
      bm[0] = part[2 * hi][col];
      bm[1] = part[2 * hi + 1][col];
      v8f cm = {};
      v8f d = __builtin_amdgcn_wmma_f32_16x16x4_f32(
          false, am, false, bm, (short)0, cm, false, false);
      const float d0 = d[0];                       // row M=0: lane n holds sum of col n
#pragma unroll
      for (int c = 0; c < C_; ++c)
        baseline[c] = bcast_lane(d0, c) * (1.0f / 128.0f);
    }
#else
#pragma unroll
    for (int c = 0; c < C_; ++c)
      baseline[c] = (part[0][c] + part[1][c] + part[2][c] + part[3][c]) * (1.0f / 128.0f);
#endif

    // ---- sequential per-contract cash / loss update (branchless) ----
#pragma unroll
    for (int c = 0; c < C_; ++c) {
      const bool oj = (new_open  >> c) & 1u;
      const bool cj = (new_close >> c) & 1u;
      const float nv = stg[4*c + 3] * cash;
      ipv[c] = oj ? nv : ipv[c];
      const float clp2 = cash_lp + (oj ? pcel[c] : 0.0f);   // cash_lp after open
      ipvl[c] = oj ? clp2 : ipvl[c];
      const float pc1 = oj ? 0.0f : pcel[c];                // pcel after open
      const float clp = ipvl[c] + pc1;
      const float cc  = cost[c];
      loss   += cj ? (clp * (cc - baseline[c]) + cc) : 0.0f;
      cash    = cash - (oj ? nv : 0.0f) + (cj ? (ipv[c] + cc) : 0.0f);
      cash_lp = clp2 + (cj ? pc1 : 0.0f);
      pcel[c] = cj ? 0.0f : pc1;
    }
  }

  out[(size_t)tid * B_ + b] = loss;   // loss shape (S, B)
}

extern "C" void kernel_launch(void* const* d_in, const int* in_sizes, int n_in,
                              void* d_out, int out_size, void* d_ws, size_t ws_size,
                              hipStream_t stream) {
  (void)in_sizes; (void)n_in; (void)out_size; (void)ws_size;
  tf_init_keys_kernel<<<1, S_, 0, stream>>>((unsigned*)d_ws);
  sim_kernel<<<B_, S_, 0, stream>>>((const float*)d_in[0], (const float*)d_in[1],
                                    (const unsigned*)d_ws, (float*)d_out);
}